// GCN_56324201120286
// MI455X (gfx1250) — compile-verified
//
#include <hip/hip_runtime.h>

typedef __attribute__((ext_vector_type(2))) float v2f;
typedef __attribute__((ext_vector_type(8))) float v8f;

#define TPB 256

// ---------------------------------------------------------------- degree ----
__global__ void k_deg_init(float* __restrict__ deg, int N) {
    int i = blockIdx.x * blockDim.x + threadIdx.x;
    if (i < N) deg[i] = 1.0f;                 // self-loop contributes 1
}

__global__ void k_deg_count(const long long* __restrict__ ei, float* __restrict__ deg,
                            int E) {
    int e = blockIdx.x * blockDim.x + threadIdx.x;
    if (e < E) {
        int d = (int)ei[(size_t)E + e];       // dst row of edge_index
        __hip_atomic_fetch_add(&deg[d], 1.0f, __ATOMIC_RELAXED, __HIP_MEMORY_SCOPE_AGENT);
    }
}

__global__ void k_dinv(float* __restrict__ deg, int N) {
    int i = blockIdx.x * blockDim.x + threadIdx.x;
    if (i < N) {
        float v = deg[i];
        deg[i] = (v > 0.0f) ? rsqrtf(v) : 0.0f;   // in place: deg -> dinv
    }
}

// ------------------------------------------------------- GEMM1: x @ W1 ------
// x [N,512] fp32, W1 [512,16] fp32 -> h1 [N,16].  One wave per 16-row tile.
// W1 staged in LDS transposed with pad 516 => bank-conflict-free float2 loads.
__global__ __launch_bounds__(TPB) void k_gemm1(const float* __restrict__ x,
                                               const float* __restrict__ W1,
                                               float* __restrict__ h1, int ntiles) {
    __shared__ float wlds[16 * 516];
    for (int idx = threadIdx.x; idx < 512 * 16; idx += TPB) {
        int k = idx >> 4, n = idx & 15;       // W1 row-major [k][n]
        wlds[n * 516 + k] = W1[idx];
    }
    __syncthreads();

    int wave = threadIdx.x >> 5;
    int lane = threadIdx.x & 31;
    int tile = blockIdx.x * (TPB / 32) + wave;
    if (tile >= ntiles) return;

    int m    = lane & 15;                     // A row / B col / D col
    int half = lane >> 4;

    const float* xrow = x + (size_t)(tile * 16 + m) * 512 + 2 * half;
    const float* wcol = &wlds[m * 516 + 2 * half];

    v8f acc = {};
#pragma unroll 4
    for (int k0 = 0; k0 < 512; k0 += 4) {
        v2f a = *(const v2f*)(xrow + k0);     // A[m][k0+2h], A[m][k0+2h+1]
        v2f b = *(const v2f*)(wcol + k0);     // B[k0+2h][n], B[k0+2h+1][n]
        acc = __builtin_amdgcn_wmma_f32_16x16x4_f32(false, a, false, b,
                                                    (short)0, acc, false, false);
    }

    float* out = h1 + (size_t)tile * 16 * 16;
#pragma unroll
    for (int v = 0; v < 8; ++v)
        out[(v + 8 * half) * 16 + m] = acc[v];   // D: M = v+8*half, N = m
}

// --------------------------------------------- self-loop init / edge scatter
__global__ void k_self_init(const float* __restrict__ h, const float* __restrict__ dinv,
                            float* __restrict__ agg, int N) {
    int i = blockIdx.x * blockDim.x + threadIdx.x;   // over N*16
    if (i < N * 16) {
        int row = i >> 4;
        float d = dinv[row];
        agg[i] = h[i] * d * d;                       // self-loop norm = dinv^2
    }
}

__global__ void k_edge_agg(const long long* __restrict__ ei, const float* __restrict__ dinv,
                           const float* __restrict__ hsrc, float* __restrict__ agg,
                           int E) {
    long long t = (long long)blockIdx.x * blockDim.x + threadIdx.x;  // E*16 threads
    if (t >= (long long)E * 16) return;
    int e = (int)(t >> 4);
    int c = (int)(t & 15);
    int s = (int)ei[e];
    int d = (int)ei[(size_t)E + e];
    float nrm = dinv[s] * dinv[d];
    float v = hsrc[(size_t)s * 16 + c] * nrm;
    __hip_atomic_fetch_add(&agg[(size_t)d * 16 + c], v,
                           __ATOMIC_RELAXED, __HIP_MEMORY_SCOPE_AGENT);
}

// bias + ReLU, and seed layer-2 self-loop aggregation (in place on agg1)
__global__ void k_relu_self2(float* __restrict__ agg1, float* __restrict__ abuf,
                             const float* __restrict__ b1, const float* __restrict__ dinv,
                             int N) {
    int i = blockIdx.x * blockDim.x + threadIdx.x;
    if (i < N * 16) {
        int row = i >> 4, c = i & 15;
        float av = fmaxf(agg1[i] + b1[c], 0.0f);
        float d  = dinv[row];
        abuf[i]  = av;                 // a = relu(agg1 + b1), gathered by edges
        agg1[i]  = av * d * d;         // becomes agg2 (self-loop seed)
    }
}

// ----------------------------------------------- GEMM2: agg2 @ W2 + b2 -----
// agg2 [N,16] @ W2 [16,40] -> z [N,40].  N-dim padded to 48 (3 WMMA tiles).
__global__ __launch_bounds__(TPB) void k_gemm2(const float* __restrict__ agg2,
                                               const float* __restrict__ W2,
                                               const float* __restrict__ b2,
                                               float* __restrict__ z, int ntiles) {
    __shared__ float w2t[48 * 20];             // col-major, pad 20 -> no conflicts
    for (int idx = threadIdx.x; idx < 48 * 20; idx += TPB) w2t[idx] = 0.0f;
    __syncthreads();
    for (int idx = threadIdx.x; idx < 16 * 40; idx += TPB) {
        int k = idx / 40, c = idx % 40;        // W2 row-major [k][c]
        w2t[c * 20 + k] = W2[idx];
    }
    __syncthreads();

    int wave = threadIdx.x >> 5;
    int lane = threadIdx.x & 31;
    int tile = blockIdx.x * (TPB / 32) + wave;
    if (tile >= ntiles) return;

    int m    = lane & 15;
    int half = lane >> 4;

    const float* arow = agg2 + (size_t)(tile * 16 + m) * 16 + 2 * half;
    v2f a0 = *(const v2f*)(arow + 0);
    v2f a1 = *(const v2f*)(arow + 4);
    v2f a2 = *(const v2f*)(arow + 8);
    v2f a3 = *(const v2f*)(arow + 12);

#pragma unroll
    for (int nt = 0; nt < 3; ++nt) {
        int col = nt * 16 + m;
        const float* bcol = &w2t[col * 20 + 2 * half];
        v2f b0 = *(const v2f*)(bcol + 0);
        v2f b1v = *(const v2f*)(bcol + 4);
        v2f b2v = *(const v2f*)(bcol + 8);
        v2f b3v = *(const v2f*)(bcol + 12);
        v8f acc = {};
        acc = __builtin_amdgcn_wmma_f32_16x16x4_f32(false, a0, false, b0,  (short)0, acc, false, false);
        acc = __builtin_amdgcn_wmma_f32_16x16x4_f32(false, a1, false, b1v, (short)0, acc, false, false);
        acc = __builtin_amdgcn_wmma_f32_16x16x4_f32(false, a2, false, b2v, (short)0, acc, false, false);
        acc = __builtin_amdgcn_wmma_f32_16x16x4_f32(false, a3, false, b3v, (short)0, acc, false, false);
        if (col < 40) {
            float bias = b2[col];
#pragma unroll
            for (int v = 0; v < 8; ++v)
                z[(size_t)(tile * 16 + v + 8 * half) * 40 + col] = acc[v] + bias;
        }
    }
}

// ------------------------------------------------------------- softmax -----
__global__ void k_softmax(const float* __restrict__ z, float* __restrict__ out, int N) {
    int i = blockIdx.x * blockDim.x + threadIdx.x;
    if (i >= N) return;
    const float* r = z + (size_t)i * 40;
    float mx = -3.402823466e38f;
    float e[40];
#pragma unroll
    for (int c = 0; c < 40; ++c) { e[c] = r[c]; mx = fmaxf(mx, e[c]); }
    float s = 0.0f;
#pragma unroll
    for (int c = 0; c < 40; ++c) { e[c] = __expf(e[c] - mx); s += e[c]; }
    float inv = 1.0f / s;
    float* o = out + (size_t)i * 40;
#pragma unroll
    for (int c = 0; c < 40; ++c) o[c] = e[c] * inv;
}

// ------------------------------------------------------------- launcher ----
extern "C" void kernel_launch(void* const* d_in, const int* in_sizes, int n_in,
                              void* d_out, int out_size, void* d_ws, size_t ws_size,
                              hipStream_t stream) {
    const float*     x   = (const float*)d_in[0];
    const long long* ei  = (const long long*)d_in[1];
    const float*     W1  = (const float*)d_in[2];
    const float*     b1  = (const float*)d_in[3];
    const float*     W2  = (const float*)d_in[4];
    const float*     b2  = (const float*)d_in[5];
    float*           out = (float*)d_out;

    const int N = in_sizes[0] / 512;          // 100000
    const int E = in_sizes[1] / 2;            // 3200000
    const int ntiles = N / 16;                // 6250 (N is a multiple of 16)

    // workspace carve-out (256B aligned): deg | bufA | bufB | z
    char* ws = (char*)d_ws;
    size_t off = 0;
    auto carve = [&](size_t bytes) { char* p = ws + off; off = (off + bytes + 255) & ~(size_t)255; return p; };
    float* deg  = (float*)carve((size_t)N * 4);        // -> dinv (in place)
    float* bufA = (float*)carve((size_t)N * 16 * 4);   // h1, later a=relu(...)
    float* bufB = (float*)carve((size_t)N * 16 * 4);   // agg1, later agg2 (in place)
    float* z    = (float*)carve((size_t)N * 40 * 4);   // pre-softmax logits

    const int gN    = (N + TPB - 1) / TPB;
    const int gN16  = (N * 16 + TPB - 1) / TPB;
    const int gE    = (E + TPB - 1) / TPB;
    const long long e16 = (long long)E * 16;
    const int gE16  = (int)((e16 + TPB - 1) / TPB);
    const int gTile = (ntiles + (TPB / 32) - 1) / (TPB / 32);

    // 1) GCN normalization: deg -> dinv
    k_deg_init<<<gN, TPB, 0, stream>>>(deg, N);
    k_deg_count<<<gE, TPB, 0, stream>>>(ei, deg, E);
    k_dinv<<<gN, TPB, 0, stream>>>(deg, N);

    // 2) layer 1: h1 = x @ W1 (WMMA), then normalized aggregation
    k_gemm1<<<gTile, TPB, 0, stream>>>(x, W1, bufA, ntiles);
    k_self_init<<<gN16, TPB, 0, stream>>>(bufA, deg, bufB, N);
    k_edge_agg<<<gE16, TPB, 0, stream>>>(ei, deg, bufA, bufB, E);

    // 3) bias+ReLU; seed layer-2 aggregation (aggregate BEFORE W2: linear map commutes)
    k_relu_self2<<<gN16, TPB, 0, stream>>>(bufB, bufA, b1, deg, N);
    k_edge_agg<<<gE16, TPB, 0, stream>>>(ei, deg, bufA, bufB, E);

    // 4) layer 2 dense transform (WMMA) + bias, then softmax
    k_gemm2<<<gTile, TPB, 0, stream>>>(bufB, W2, b2, z, ntiles);
    k_softmax<<<gN, TPB, 0, stream>>>(z, out, N);
}